// TrajClassifier_29970281791549
// MI455X (gfx1250) — compile-verified
//
#include <hip/hip_runtime.h>

// ---------------------------------------------------------------------------
// TrajClassifier forward for gfx1250 (MI455X). fp32 everywhere; all GEMMs go
// through V_WMMA_F32_16X16X4_F32. One wave computes a 32x32 output tile
// (2x2 subtiles, 4 accumulators); the all-interior case (everything except the
// final N=1 layer) takes a branch-free loop with no EXEC masking.
// ---------------------------------------------------------------------------

typedef float v2f __attribute__((ext_vector_type(2)));
typedef float v8f __attribute__((ext_vector_type(8)));

#define LSLOPE 0.01f

// ------------------------------- kernels -----------------------------------

__global__ void k_fill(float* __restrict__ p, float v, int n) {
  int i = blockIdx.x * blockDim.x + threadIdx.x;
  if (i < n) p[i] = v;
}

// Y = leaky(X @ W^T + b), tiny K (embeds: K = 2/3)
__global__ void k_linear_small(const float* __restrict__ X, const float* __restrict__ W,
                               const float* __restrict__ b, float* __restrict__ Y,
                               int n, int din, int dout) {
  int idx = blockIdx.x * blockDim.x + threadIdx.x;
  if (idx >= n * dout) return;
  int r = idx / dout, o = idx - r * dout;
  const float* xr = X + (long)r * din;
  const float* wr = W + (long)o * din;
  float acc = b[o];
  for (int k = 0; k < din; ++k) acc += xr[k] * wr[k];
  Y[idx] = acc > 0.f ? acc : LSLOPE * acc;
}

// Y[M,N] = X[M,K](ldx) @ W[N,K]^T (+bias) ; flags bit1: Y += result ; bit0: leaky
// One wave computes a 32x32 tile as 2x2 16x16 WMMA subtiles.
// A frag: lanes 0-15 hold row m+l, VGPR pair = K {0,1}; lanes 16-31 = K {2,3}.
// B frag mirrors A over W rows (columns of W^T). C/D: ISA 16x16 f32 layout.
__global__ void k_gemm_wmma(const float* __restrict__ X, int ldx,
                            const float* __restrict__ W,
                            const float* __restrict__ bias,
                            float* __restrict__ Y, int ldy,
                            int M, int N, int K, int flags) {
  const int m0 = blockIdx.x << 5;
  const int n0 = blockIdx.y << 5;
  const int lane = threadIdx.x;
  const int half = lane >> 4;
  const int l16 = lane & 15;
  v8f acc00 = {}, acc01 = {}, acc10 = {}, acc11 = {};

  if (m0 + 32 <= M && n0 + 32 <= N) {
    // interior tile: uniform branch, no EXEC masking in the hot loop
    const float* __restrict__ xp0 = X + (long)(m0 + l16) * ldx + 2 * half;
    const float* __restrict__ xp1 = X + (long)(m0 + 16 + l16) * ldx + 2 * half;
    const float* __restrict__ wp0 = W + (long)(n0 + l16) * K + 2 * half;
    const float* __restrict__ wp1 = W + (long)(n0 + 16 + l16) * K + 2 * half;
    for (int k0 = 0; k0 < K; k0 += 4) {
      v2f a0 = *(const v2f*)(xp0 + k0);
      v2f a1 = *(const v2f*)(xp1 + k0);
      v2f b0 = *(const v2f*)(wp0 + k0);
      v2f b1 = *(const v2f*)(wp1 + k0);
      acc00 = __builtin_amdgcn_wmma_f32_16x16x4_f32(false, a0, false, b0, (short)0, acc00,
                                                    false, false);
      acc01 = __builtin_amdgcn_wmma_f32_16x16x4_f32(false, a0, false, b1, (short)0, acc01,
                                                    false, false);
      acc10 = __builtin_amdgcn_wmma_f32_16x16x4_f32(false, a1, false, b0, (short)0, acc10,
                                                    false, false);
      acc11 = __builtin_amdgcn_wmma_f32_16x16x4_f32(false, a1, false, b1, (short)0, acc11,
                                                    false, false);
    }
  } else {
    // edge tile: per-lane guarded loads (only the N=1 head layer hits this)
    const int ar0 = m0 + l16, ar1 = m0 + 16 + l16;
    const int br0 = n0 + l16, br1 = n0 + 16 + l16;
    const bool a0ok = ar0 < M, a1ok = ar1 < M, b0ok = br0 < N, b1ok = br1 < N;
    for (int k0 = 0; k0 < K; k0 += 4) {
      const int ka = k0 + 2 * half;
      v2f a0 = {0.f, 0.f}, a1 = {0.f, 0.f}, b0 = {0.f, 0.f}, b1 = {0.f, 0.f};
      if (a0ok) { a0.x = X[(long)ar0 * ldx + ka]; a0.y = X[(long)ar0 * ldx + ka + 1]; }
      if (a1ok) { a1.x = X[(long)ar1 * ldx + ka]; a1.y = X[(long)ar1 * ldx + ka + 1]; }
      if (b0ok) { b0.x = W[(long)br0 * K + ka];   b0.y = W[(long)br0 * K + ka + 1]; }
      if (b1ok) { b1.x = W[(long)br1 * K + ka];   b1.y = W[(long)br1 * K + ka + 1]; }
      acc00 = __builtin_amdgcn_wmma_f32_16x16x4_f32(false, a0, false, b0, (short)0, acc00,
                                                    false, false);
      acc01 = __builtin_amdgcn_wmma_f32_16x16x4_f32(false, a0, false, b1, (short)0, acc01,
                                                    false, false);
      acc10 = __builtin_amdgcn_wmma_f32_16x16x4_f32(false, a1, false, b0, (short)0, acc10,
                                                    false, false);
      acc11 = __builtin_amdgcn_wmma_f32_16x16x4_f32(false, a1, false, b1, (short)0, acc11,
                                                    false, false);
    }
  }

  auto store = [&](const v8f& acc, int rbase, int col) {
    if (col >= N) return;
    const float bv = bias ? bias[col] : 0.f;
#pragma unroll
    for (int v = 0; v < 8; ++v) {
      const int row = rbase + v;
      if (row < M) {
        float r = acc[v] + bv;
        if (flags & 2) r += Y[(long)row * ldy + col];
        if (flags & 1) r = r > 0.f ? r : LSLOPE * r;
        Y[(long)row * ldy + col] = r;
      }
    }
  };
  const int col0 = n0 + l16, col1 = n0 + 16 + l16;
  const int rb0 = m0 + (half << 3), rb1 = m0 + 16 + (half << 3);
  store(acc00, rb0, col0);
  store(acc01, rb0, col1);
  store(acc10, rb1, col0);
  store(acc11, rb1, col1);
}

// gates[n,4H] already holds e_t@Wih^T + h@Whh^T; add biases, apply cell.
// torch gate order i,f,g,o. Optionally store h into hs (lane bi-LSTM seq).
__global__ void k_lstm_cell(const float* __restrict__ gates,
                            const float* __restrict__ bih, const float* __restrict__ bhh,
                            float* __restrict__ h, float* __restrict__ c,
                            float* __restrict__ hs, int hs_row_stride, int hs_col_off,
                            int n, int H) {
  int idx = blockIdx.x * blockDim.x + threadIdx.x;
  if (idx >= n * H) return;
  int row = idx / H, j = idx - row * H;
  const float* g = gates + (long)row * 4 * H;
  float gi = g[j]         + bih[j]         + bhh[j];
  float gf = g[H + j]     + bih[H + j]     + bhh[H + j];
  float gg = g[2 * H + j] + bih[2 * H + j] + bhh[2 * H + j];
  float go = g[3 * H + j] + bih[3 * H + j] + bhh[3 * H + j];
  float i_ = 1.f / (1.f + expf(-gi));
  float f_ = 1.f / (1.f + expf(-gf));
  float g_ = tanhf(gg);
  float o_ = 1.f / (1.f + expf(-go));
  float cv = f_ * c[idx] + i_ * g_;
  float hv = o_ * tanhf(cv);
  c[idx] = cv;
  h[idx] = hv;
  if (hs) hs[(long)row * hs_row_stride + hs_col_off + j] = hv;
}

// Y[r, 0:da] = A[(r/divA)*mulA + r%modA, :da] ; Y[r, da:da+db] = B[r, :db]
__global__ void k_concat2(const float* __restrict__ A, int lda, int divA, int mulA, int modA,
                          const float* __restrict__ Bp, int ldb,
                          float* __restrict__ Y, int n, int da, int db) {
  int d = da + db;
  int idx = blockIdx.x * blockDim.x + threadIdx.x;
  if (idx >= n * d) return;
  int r = idx / d, col = idx - r * d;
  float v;
  if (col < da) {
    int ar = (r / divA) * mulA + (r % modA);
    v = A[(long)ar * lda + col];
  } else {
    v = Bp[(long)r * ldb + (col - da)];
  }
  Y[idx] = v;
}

// nbr xy input: [1024*20] rows of (x, y, padding)
__global__ void k_build_nbr_in(const float* __restrict__ xy, const float* __restrict__ pad,
                               float* __restrict__ Y, int n) {
  int idx = blockIdx.x * blockDim.x + threadIdx.x;
  if (idx >= n) return;
  Y[idx * 3 + 0] = xy[idx * 2 + 0];
  Y[idx * 3 + 1] = xy[idx * 2 + 1];
  Y[idx * 3 + 2] = pad[idx];
}

// Y[groups, nn+1, d]: first nn rows from nbr[g*nn+j], last row obs[g]
__global__ void k_build_allobs(const float* __restrict__ nbr, const float* __restrict__ obs,
                               float* __restrict__ Y, int groups, int nn, int d) {
  int total = groups * (nn + 1) * d;
  int idx = blockIdx.x * blockDim.x + threadIdx.x;
  if (idx >= total) return;
  int g = idx / ((nn + 1) * d);
  int rem = idx - g * (nn + 1) * d;
  int j = rem / d, col = rem - j * d;
  Y[idx] = (j < nn) ? nbr[((long)g * nn + j) * d + col] : obs[(long)g * d + col];
}

// batched scaled-dot attention context: ctx[g,q,:] = softmax(qk.k^T/sqrt(dk)).v
// mode 0: kv group = g ; mode 1: g = b*64+cp -> kv group = b*8 + cp/8
__global__ void k_attn(const float* __restrict__ qk, const float* __restrict__ kmat,
                       const float* __restrict__ vmat, float* __restrict__ ctx,
                       int Q, int Kn, int dk, int dv, int mode) {
  __shared__ float qv[128];
  __shared__ float s[64];
  __shared__ float red[2];
  int bq = blockIdx.x;          // g*Q + q
  int g = bq / Q;
  int kvg = (mode == 0) ? g : ((g >> 6) * 8 + ((g & 63) >> 3));
  int t = threadIdx.x;          // 0..127
  if (t < dk) qv[t] = qk[(long)bq * dk + t];
  __syncthreads();
  float scale = rsqrtf((float)dk);
  if (t < Kn) {
    const float* kr = kmat + ((long)kvg * Kn + t) * dk;
    float acc = 0.f;
    for (int i = 0; i < dk; ++i) acc += qv[i] * kr[i];
    s[t] = acc * scale;
  }
  __syncthreads();
  if (t == 0) {
    float mx = -1e30f;
    for (int i = 0; i < Kn; ++i) mx = fmaxf(mx, s[i]);
    red[0] = mx;
  }
  __syncthreads();
  float mx = red[0];
  if (t < Kn) s[t] = expf(s[t] - mx);
  __syncthreads();
  if (t == 0) {
    float sm = 0.f;
    for (int i = 0; i < Kn; ++i) sm += s[i];
    red[1] = sm;
  }
  __syncthreads();
  float inv = 1.f / red[1];
  if (t < dv) {
    float acc = 0.f;
    for (int k2 = 0; k2 < Kn; ++k2)
      acc += s[k2] * vmat[((long)kvg * Kn + k2) * dv + t];
    ctx[(long)bq * dv + t] = acc * inv;
  }
}

// combin[r, :] = [allobs_lane'[(b*8+c)*33+32], social'[same], allpreds[r]]
__global__ void k_build_combin(const float* __restrict__ aol, const float* __restrict__ soc,
                               const float* __restrict__ ap, float* __restrict__ Y) {
  int idx = blockIdx.x * blockDim.x + threadIdx.x;
  if (idx >= 2048 * 384) return;
  int r = idx / 384, col = idx - r * 384;
  int b = r >> 6, cp = r & 63, c = cp >> 3;
  long gc = ((long)(b * 8 + c) * 33 + 32) * 128;
  float v;
  if (col < 128)      v = aol[gc + col];
  else if (col < 256) v = soc[gc + (col - 128)];
  else                v = ap[(long)r * 128 + (col - 256)];
  Y[idx] = v;
}

// ------------------------------- host side ---------------------------------

// param leaf indices (params dict tree-flattened with sorted keys, after the
// 8 data inputs; uppercase sorts before lowercase)
enum {
  P_ALLP_W = 8, P_ALLP_B, P_CLSMAP_W, P_CLSMAP_B,
  P_F2F_W, P_F2F_B, P_H2H_W, P_H2H_B,
  P_L2F_W, P_L2F_B, P_L2H_W, P_L2H_B,
  P_LANE_B_WHH, P_LANE_B_WIH, P_LANE_B_BHH, P_LANE_B_BIH,   // 20..23
  P_LANE_EW, P_LANE_EB,                                     // 24,25
  P_LANE_F_WHH, P_LANE_F_WIH, P_LANE_F_BHH, P_LANE_F_BIH,   // 26..29
  P_NBR_ENC_W = 30, P_NBR_ENC_B, P_NBR_MAP_W, P_NBR_MAP_B,
  P_NBR_SD = 34,              // Whh,Wih,bhh,bih,eW,eb
  P_NBR_XY = 40,
  P_OBS_ENC_W = 46, P_OBS_ENC_B, P_OBS_MAP_W, P_OBS_MAP_B,
  P_OBS_SD = 50,
  P_OBS_XY = 56,
  P_PRED_ENC_W = 62, P_PRED_ENC_B, P_PRED_MAP_W, P_PRED_MAP_B,
  P_PRED_SD = 66,
  P_PRED_XY = 72,
  P_SD1_W = 78, P_SD1_B, P_SD2_W, P_SD2_B, P_SD3_W, P_SD3_B,
  P_SOCIAL_W = 84, P_SOCIAL_B
};

extern "C" void kernel_launch(void* const* d_in, const int* in_sizes, int n_in,
                              void* d_out, int out_size, void* d_ws, size_t ws_size,
                              hipStream_t stream) {
  (void)in_sizes; (void)n_in; (void)out_size; (void)ws_size;

  const float* agent_cls     = (const float*)d_in[0];  // [256,64,2]
  const float* agent_obs_xy  = (const float*)d_in[1];  // [32,20,2]
  const float* agent_futs_xy = (const float*)d_in[2];  // [2048,30,2]
  const float* nbrs_obs_xy   = (const float*)d_in[3];  // [1024,20,2]
  const float* nbrs_pad      = (const float*)d_in[4];  // [1024,20]
  const float* agent_obs_sds = (const float*)d_in[5];  // [256,20,2]
  const float* agent_futs_sd = (const float*)d_in[6];  // [2048,30,2]
  const float* nbrs_obs_sds  = (const float*)d_in[7];  // [8192,20,2]
  auto PA = [&](int i) { return (const float*)d_in[i]; };

  // workspace allocator (floats, 256B aligned chunks)
  float* wbase = (float*)d_ws;
  size_t off = 0;
  auto alloc = [&](size_t nf) { float* p = wbase + off; off += (nf + 63) & ~(size_t)63; return p; };

  float* lane_e     = alloc(16384 * 16);
  float* cls_enc    = alloc(16384 * 64);   // [256,64,64]  concat(hf,hb)
  float* cls_mapb   = alloc(16384 * 64);
  float* lane_h     = alloc(256 * 32);
  float* lane_c     = alloc(256 * 32);
  float* lane_gates = alloc(256 * 128);
  float* ebuf       = alloc(163840 * 32);  // max n*T*32 (nbr_sd)
  float* gbuf       = alloc(8192 * 512);   // max n*4H
  float* cbuf       = alloc(8192 * 128);
  float* nbr_in3    = alloc(1024 * 20 * 3);
  float* obs_xy_h   = alloc(32 * 128);
  float* obs_sd_h   = alloc(256 * 128);
  float* nbr_xy_h   = alloc(1024 * 128);
  float* nbr_sd_h   = alloc(8192 * 128);
  float* pr_xy_h    = alloc(2048 * 128);
  float* pr_sd_h    = alloc(2048 * 128);
  float* cat        = alloc(8192 * 256);   // shared concat scratch
  float* obs_enc    = alloc(256 * 128);
  float* obs_map    = alloc(256 * 64);
  float* nbr_enc    = alloc(8192 * 128);
  float* nbr_map    = alloc(8192 * 64);
  float* pr_encb    = alloc(2048 * 128);
  float* pr_mapb    = alloc(2048 * 64);
  float* allobs_e   = alloc(256 * 33 * 128);
  float* allobs_m   = alloc(256 * 33 * 64);
  float* ctx1       = alloc(256 * 33 * 64);
  float* allobs_ln  = alloc(256 * 33 * 128);
  float* soc_map    = alloc(256 * 33 * 128);
  float* ctx2       = alloc(256 * 33 * 128);
  float* social_b   = alloc(256 * 33 * 128);
  float* ctx3       = alloc(2048 * 64);
  float* pred_lane  = alloc(2048 * 128);
  float* pl_mapb    = alloc(2048 * 128);
  float* ctx4       = alloc(2048 * 128);
  float* allpreds   = alloc(2048 * 128);
  float* combin     = alloc(2048 * 384);
  float* mlp1       = alloc(2048 * 384);
  float* mlp2       = alloc(2048 * 384);

  // launch helpers
  auto fill = [&](float* p, float v, long n) {
    k_fill<<<dim3((unsigned)((n + 255) / 256)), dim3(256), 0, stream>>>(p, v, (int)n);
  };
  auto gemm = [&](const float* X, int ldx, const float* W, const float* bias,
                  float* Y, int ldy, int M, int N, int K, int flags) {
    dim3 g((M + 31) / 32, (N + 31) / 32);
    k_gemm_wmma<<<g, dim3(32), 0, stream>>>(X, ldx, W, bias, Y, ldy, M, N, K, flags);
  };
  auto cell = [&](const float* g, const float* bih, const float* bhh, float* h, float* c,
                  float* hs, int hstride, int hoff, int n, int H) {
    int tot = n * H;
    k_lstm_cell<<<dim3((tot + 255) / 256), dim3(256), 0, stream>>>(
        g, bih, bhh, h, c, hs, hstride, hoff, n, H);
  };
  auto lin_small = [&](const float* X, const float* W, const float* b, float* Y,
                       int n, int din, int dout) {
    int tot = n * dout;
    k_linear_small<<<dim3((tot + 255) / 256), dim3(256), 0, stream>>>(X, W, b, Y, n, din, dout);
  };
  auto concat2 = [&](const float* A, int lda, int divA, int mulA, int modA,
                     const float* B, int ldb, float* Y, int n, int da, int db) {
    int tot = n * (da + db);
    k_concat2<<<dim3((tot + 255) / 256), dim3(256), 0, stream>>>(
        A, lda, divA, mulA, modA, B, ldb, Y, n, da, db);
  };
  auto attn = [&](const float* qk, const float* kk, const float* vv, float* ctx,
                  int G, int Q, int Kn, int dk, int dv, int mode) {
    k_attn<<<dim3(G * Q), dim3(128), 0, stream>>>(qk, kk, vv, ctx, Q, Kn, dk, dv, mode);
  };

  // ---- lane encoder: embed + bi-LSTM (H=32) over L=64 ----
  lin_small(agent_cls, PA(P_LANE_EW), PA(P_LANE_EB), lane_e, 16384, 2, 16);
  fill(lane_h, 0.f, 256 * 32);
  fill(lane_c, 0.f, 256 * 32);
  for (int t = 0; t < 64; ++t) {
    gemm(lane_e + t * 16, 64 * 16, PA(P_LANE_F_WIH), nullptr, lane_gates, 128, 256, 128, 16, 0);
    gemm(lane_h, 32, PA(P_LANE_F_WHH), nullptr, lane_gates, 128, 256, 128, 32, 2);
    cell(lane_gates, PA(P_LANE_F_BIH), PA(P_LANE_F_BHH), lane_h, lane_c,
         cls_enc + t * 64, 64 * 64, 0, 256, 32);
  }
  fill(lane_h, 0.f, 256 * 32);
  fill(lane_c, 0.f, 256 * 32);
  for (int s = 0; s < 64; ++s) {
    int t = 63 - s;
    gemm(lane_e + t * 16, 64 * 16, PA(P_LANE_B_WIH), nullptr, lane_gates, 128, 256, 128, 16, 0);
    gemm(lane_h, 32, PA(P_LANE_B_WHH), nullptr, lane_gates, 128, 256, 128, 32, 2);
    cell(lane_gates, PA(P_LANE_B_BIH), PA(P_LANE_B_BHH), lane_h, lane_c,
         cls_enc + t * 64, 64 * 64, 32, 256, 32);
  }
  gemm(cls_enc, 64, PA(P_CLSMAP_W), PA(P_CLSMAP_B), cls_mapb, 64, 16384, 64, 64, 1);

  // ---- trajectory encoders: embed(32) + LSTM(H=128), keep last h ----
  auto traj_enc = [&](const float* x, int n, int T, int din, int pbase, float* hout) {
    lin_small(x, PA(pbase + 4), PA(pbase + 5), ebuf, n * T, din, 32);
    fill(hout, 0.f, (long)n * 128);
    fill(cbuf, 0.f, (long)n * 128);
    for (int t = 0; t < T; ++t) {
      gemm(ebuf + t * 32, T * 32, PA(pbase + 1), nullptr, gbuf, 512, n, 512, 32, 0);
      gemm(hout, 128, PA(pbase + 0), nullptr, gbuf, 512, n, 512, 128, 2);
      cell(gbuf, PA(pbase + 3), PA(pbase + 2), hout, cbuf, nullptr, 0, 0, n, 128);
    }
  };
  traj_enc(agent_obs_xy, 32, 20, 2, P_OBS_XY, obs_xy_h);
  traj_enc(agent_obs_sds, 256, 20, 2, P_OBS_SD, obs_sd_h);
  {
    int tot = 1024 * 20;
    k_build_nbr_in<<<dim3((tot + 255) / 256), dim3(256), 0, stream>>>(nbrs_obs_xy, nbrs_pad,
                                                                      nbr_in3, tot);
  }
  traj_enc(nbr_in3, 1024, 20, 3, P_NBR_XY, nbr_xy_h);
  traj_enc(nbrs_obs_sds, 8192, 20, 2, P_NBR_SD, nbr_sd_h);
  traj_enc(agent_futs_xy, 2048, 30, 2, P_PRED_XY, pr_xy_h);
  traj_enc(agent_futs_sd, 2048, 30, 2, P_PRED_SD, pr_sd_h);

  // ---- fused encodings ----
  // obs: concat(repeat(obs_xy_h, C), obs_sd_h) -> enc -> map
  concat2(obs_xy_h, 128, 8, 1, 1, obs_sd_h, 128, cat, 256, 128, 128);
  gemm(cat, 256, PA(P_OBS_ENC_W), PA(P_OBS_ENC_B), obs_enc, 128, 256, 128, 256, 1);
  gemm(obs_enc, 128, PA(P_OBS_MAP_W), PA(P_OBS_MAP_B), obs_map, 64, 256, 64, 128, 1);
  // nbr: rows (b,c,n); xy rep row = b*32+n
  concat2(nbr_xy_h, 128, 256, 32, 32, nbr_sd_h, 128, cat, 8192, 128, 128);
  gemm(cat, 256, PA(P_NBR_ENC_W), PA(P_NBR_ENC_B), nbr_enc, 128, 8192, 128, 256, 1);
  gemm(nbr_enc, 128, PA(P_NBR_MAP_W), PA(P_NBR_MAP_B), nbr_map, 64, 8192, 64, 128, 1);
  // pred
  concat2(pr_xy_h, 128, 1, 1, 1, pr_sd_h, 128, cat, 2048, 128, 128);
  gemm(cat, 256, PA(P_PRED_ENC_W), PA(P_PRED_ENC_B), pr_encb, 128, 2048, 128, 256, 1);
  gemm(pr_encb, 128, PA(P_PRED_MAP_W), PA(P_PRED_MAP_B), pr_mapb, 64, 2048, 64, 128, 1);

  // ---- attention stack (L2H/H2H at [B,C,33,.]; p-invariant, gathered later) ----
  {
    int tot = 256 * 33 * 128;
    k_build_allobs<<<dim3((tot + 255) / 256), dim3(256), 0, stream>>>(nbr_enc, obs_enc,
                                                                      allobs_e, 256, 32, 128);
    tot = 256 * 33 * 64;
    k_build_allobs<<<dim3((tot + 255) / 256), dim3(256), 0, stream>>>(nbr_map, obs_map,
                                                                      allobs_m, 256, 32, 64);
  }
  // L2H: actors attend to candidate lane
  attn(allobs_m, cls_mapb, cls_enc, ctx1, 256, 33, 64, 64, 64, 0);
  concat2(allobs_e, 128, 1, 1, 1, ctx1, 64, cat, 256 * 33, 128, 64);
  gemm(cat, 192, PA(P_L2H_W), PA(P_L2H_B), allobs_ln, 128, 256 * 33, 128, 192, 1);
  // H2H
  gemm(allobs_ln, 128, PA(P_SOCIAL_W), PA(P_SOCIAL_B), soc_map, 128, 256 * 33, 128, 128, 1);
  attn(soc_map, soc_map, allobs_ln, ctx2, 256, 33, 33, 128, 128, 0);
  concat2(allobs_ln, 128, 1, 1, 1, ctx2, 128, cat, 256 * 33, 128, 128);
  gemm(cat, 256, PA(P_H2H_W), PA(P_H2H_B), social_b, 128, 256 * 33, 128, 256, 1);
  // L2F: each prediction attends to its lane (kv group = b*8 + cp/8)
  attn(pr_mapb, cls_mapb, cls_enc, ctx3, 2048, 1, 64, 64, 64, 1);
  concat2(pr_encb, 128, 1, 1, 1, ctx3, 64, cat, 2048, 128, 64);
  gemm(cat, 192, PA(P_L2F_W), PA(P_L2F_B), pred_lane, 128, 2048, 128, 192, 1);
  // F2F: predictions attend to each other within a sample
  gemm(pred_lane, 128, PA(P_ALLP_W), PA(P_ALLP_B), pl_mapb, 128, 2048, 128, 128, 1);
  attn(pl_mapb, pl_mapb, pred_lane, ctx4, 32, 64, 64, 128, 128, 0);
  concat2(pred_lane, 128, 1, 1, 1, ctx4, 128, cat, 2048, 128, 128);
  gemm(cat, 256, PA(P_F2F_W), PA(P_F2F_B), allpreds, 128, 2048, 128, 256, 1);

  // ---- head: gather combin, 384-384-384-1 MLP ----
  {
    int tot = 2048 * 384;
    k_build_combin<<<dim3((tot + 255) / 256), dim3(256), 0, stream>>>(allobs_ln, social_b,
                                                                      allpreds, combin);
  }
  gemm(combin, 384, PA(P_SD1_W), PA(P_SD1_B), mlp1, 384, 2048, 384, 384, 1);
  gemm(mlp1, 384, PA(P_SD2_W), PA(P_SD2_B), mlp2, 384, 2048, 384, 384, 1);
  gemm(mlp2, 384, PA(P_SD3_W), PA(P_SD3_B), (float*)d_out, 1, 2048, 1, 384, 0);
}